// CRD_30459908063689
// MI455X (gfx1250) — compile-verified
//
#include <hip/hip_runtime.h>
#include <hip/hip_bf16.h>

typedef float v2f __attribute__((ext_vector_type(2)));
typedef float v8f __attribute__((ext_vector_type(8)));

#define D_IN 128
#define D_OUT 32

// ---------------------------------------------------------------------------
// 1) init: deg = 1.0 (self loop), agg = 0
// ---------------------------------------------------------------------------
__global__ void init_kernel(float* __restrict__ deg, float* __restrict__ agg,
                            int N, int totalAgg) {
    int i = blockIdx.x * blockDim.x + threadIdx.x;
    if (i < totalAgg) agg[i] = 0.0f;
    if (i < N) deg[i] = 1.0f;
}

// ---------------------------------------------------------------------------
// 2) in-degree accumulation over dst (f32 atomics, exact for these counts)
// ---------------------------------------------------------------------------
__global__ void degree_kernel(const long long* __restrict__ dst,
                              float* __restrict__ deg, int E) {
    int e = blockIdx.x * blockDim.x + threadIdx.x;
    if (e < E) {
        unsafeAtomicAdd(&deg[(int)dst[e]], 1.0f);
    }
}

// ---------------------------------------------------------------------------
// 3) deg <- rsqrt(deg)   (deg >= 1 always, so no zero guard needed)
// ---------------------------------------------------------------------------
__global__ void rsqrt_kernel(float* __restrict__ deg, int N) {
    int i = blockIdx.x * blockDim.x + threadIdx.x;
    if (i < N) deg[i] = rsqrtf(deg[i]);
}

// ---------------------------------------------------------------------------
// 4) h = X @ W via v_wmma_f32_16x16x4_f32.
//    One wave computes a 16x32 output strip (two 16x16 C tiles).
//    W (128x32 = 16KB) staged in LDS once per block.
// ---------------------------------------------------------------------------
__global__ __launch_bounds__(256) void gemm_wmma_kernel(
    const float* __restrict__ x, const float* __restrict__ W,
    float* __restrict__ h, int N) {
    __shared__ float ldsW[D_IN * D_OUT];  // 16 KB of the 320 KB/WGP LDS

    // cooperative load of W into LDS (4096 floats, 256 threads -> 16 each)
    for (int i = threadIdx.x; i < D_IN * D_OUT; i += blockDim.x)
        ldsW[i] = W[i];
    __syncthreads();

    const int wavesPerBlock = blockDim.x >> 5;
    const int wave  = blockIdx.x * wavesPerBlock + (threadIdx.x >> 5);
    const int row0  = wave * 16;
    if (row0 >= N) return;  // wave-uniform: N % 16 == 0, EXEC stays all-ones

    const int lane   = threadIdx.x & 31;
    const int laneLo = lane & 15;
    const int laneHi = lane >> 4;  // 0 or 1

    v8f c0 = {};  // cols 0..15
    v8f c1 = {};  // cols 16..31

    const float* xrow = x + (size_t)(row0 + laneLo) * D_IN;

    #pragma unroll 4
    for (int k = 0; k < D_IN; k += 4) {
        // A 16x4 f32 frag: a[v] = X[row0+laneLo][k + v + 2*laneHi]
        v2f a = *(const v2f*)(xrow + k + 2 * laneHi);

        // B 4x16 f32 frags: b[v] = W[k + v + 2*laneHi][colBase + laneLo]
        const float* wp = ldsW + (size_t)(k + 2 * laneHi) * D_OUT + laneLo;
        v2f b0, b1;
        b0.x = wp[0];          // K = k + 2*laneHi,     cols 0..15
        b0.y = wp[D_OUT];      // K = k + 2*laneHi + 1
        b1.x = wp[16];         // cols 16..31
        b1.y = wp[D_OUT + 16];

        c0 = __builtin_amdgcn_wmma_f32_16x16x4_f32(false, a, false, b0,
                                                   (short)0, c0, false, false);
        c1 = __builtin_amdgcn_wmma_f32_16x16x4_f32(false, a, false, b1,
                                                   (short)0, c1, false, false);
    }

    // C/D layout: element v -> row (v + 8*laneHi), col laneLo
    float* hp = h + (size_t)row0 * D_OUT;
    #pragma unroll
    for (int v = 0; v < 8; ++v) {
        hp[(size_t)(v + 8 * laneHi) * D_OUT + laneLo]       = c0[v];
        hp[(size_t)(v + 8 * laneHi) * D_OUT + 16 + laneLo]  = c1[v];
    }
}

// ---------------------------------------------------------------------------
// 5) edge scatter: agg[dst] += dinv[src]*dinv[dst] * h[src]
//    8 threads per edge, float4 each; atomics land in the 192MB L2
//    since agg (12.8MB) is L2-resident.
// ---------------------------------------------------------------------------
__global__ void scatter_kernel(const long long* __restrict__ src,
                               const long long* __restrict__ dst,
                               const float* __restrict__ dinv,
                               const float* __restrict__ h,
                               float* __restrict__ agg, int E) {
    int t = blockIdx.x * blockDim.x + threadIdx.x;
    int e = t >> 3;
    if (e >= E) return;
    int q = t & 7;

    int s = (int)src[e];
    int d = (int)dst[e];
    float w = dinv[s] * dinv[d];

    const float4 hv = *(const float4*)(h + (size_t)s * D_OUT + q * 4);
    float* ap = agg + (size_t)d * D_OUT + q * 4;
    unsafeAtomicAdd(ap + 0, w * hv.x);
    unsafeAtomicAdd(ap + 1, w * hv.y);
    unsafeAtomicAdd(ap + 2, w * hv.z);
    unsafeAtomicAdd(ap + 3, w * hv.w);
}

// ---------------------------------------------------------------------------
// 6) finalize: out = relu(agg + dinv^2 * h + b)  (self-loop folded in here)
// ---------------------------------------------------------------------------
__global__ void finalize_kernel(const float* __restrict__ agg,
                                const float* __restrict__ h,
                                const float* __restrict__ dinv,
                                const float* __restrict__ b,
                                float* __restrict__ out, int total) {
    int i = blockIdx.x * blockDim.x + threadIdx.x;
    if (i >= total) return;
    int node = i >> 5;  // D_OUT == 32
    int j    = i & 31;
    float di = dinv[node];
    float v  = agg[i] + di * di * h[i] + b[j];
    out[i] = fmaxf(v, 0.0f);
}

// ---------------------------------------------------------------------------
extern "C" void kernel_launch(void* const* d_in, const int* in_sizes, int n_in,
                              void* d_out, int out_size, void* d_ws, size_t ws_size,
                              hipStream_t stream) {
    const float*     x  = (const float*)d_in[0];
    const long long* ei = (const long long*)d_in[1];  // int64 edge_index [2, E]
    const float*     W  = (const float*)d_in[2];
    const float*     b  = (const float*)d_in[3];
    float*           out = (float*)d_out;

    const int N = in_sizes[0] / D_IN;   // 100000
    const int E = in_sizes[1] / 2;      // 1600000
    const long long* src = ei;
    const long long* dst = ei + E;

    // workspace layout: deg/dinv [N] | h [N*32] | agg [N*32]
    float* deg = (float*)d_ws;
    float* h   = deg + N;
    float* agg = h + (size_t)N * D_OUT;

    const int totalFeat = N * D_OUT;

    // 1) init deg=1, agg=0
    {
        int threads = 256, blocks = (totalFeat + threads - 1) / threads;
        init_kernel<<<blocks, threads, 0, stream>>>(deg, agg, N, totalFeat);
    }
    // 2) degree atomics
    {
        int threads = 256, blocks = (E + threads - 1) / threads;
        degree_kernel<<<blocks, threads, 0, stream>>>(dst, deg, E);
    }
    // 3) deg -> rsqrt(deg)
    {
        int threads = 256, blocks = (N + threads - 1) / threads;
        rsqrt_kernel<<<blocks, threads, 0, stream>>>(deg, N);
    }
    // 4) h = X @ W  (WMMA)
    {
        int threads = 256;
        int wavesPerBlock = threads / 32;
        int nTiles = (N + 15) / 16;
        int blocks = (nTiles + wavesPerBlock - 1) / wavesPerBlock;
        gemm_wmma_kernel<<<blocks, threads, 0, stream>>>(x, W, h, N);
    }
    // 5) edge scatter
    {
        int threads = 256;
        long long totalT = (long long)E * 8;
        int blocks = (int)((totalT + threads - 1) / threads);
        scatter_kernel<<<blocks, threads, 0, stream>>>(src, dst, deg, h, agg, E);
    }
    // 6) finalize
    {
        int threads = 256, blocks = (totalFeat + threads - 1) / threads;
        finalize_kernel<<<blocks, threads, 0, stream>>>(agg, h, deg, b, out, totalFeat);
    }
}